// GaussianKernel_28862180229634
// MI455X (gfx1250) — compile-verified
//
#include <hip/hip_runtime.h>

// Problem constants (match reference)
#define B_  8
#define K_  288
#define L_  16384
#define O_  64

// Tiling
#define BLOCK_L 64            // l-tile per workgroup
#define KT      96            // K-tile staged in LDS (3 tiles cover K=288)
#define K2T     (KT / 2)      // 48 pair-rows per tile
#define PAIRROW 160           // words per pair-row: 64 cols * 2 (k,k+1 interleave) + 32 pad
                              // 160 mod 64 == 32 -> half-wave lands on opposite bank group
#define OROW    68            // padded row stride (words) for output repack staging
#define NTHREADS 256          // 8 waves (wave32)
#define UNITS   ((K2T * 16) / NTHREADS)   // 3 row-pair units per thread per tile

typedef __attribute__((ext_vector_type(2))) float v2f;
typedef __attribute__((ext_vector_type(8))) float v8f;

__global__ __launch_bounds__(NTHREADS)
void gaussian_wmma_f32_kernel(const float* __restrict__ x,     // [B, K, L]
                              const float* __restrict__ w,     // [K, O] (memory reinterpret of [O,32,3,3])
                              const float* __restrict__ bias,  // [O]
                              const float* __restrict__ gamma, // [1]
                              float* __restrict__ out)         // [B, L, O]
{
    __shared__ float sW[K2T * PAIRROW];   // 30720 B: W tile, pair-interleaved [k2][o][2]
    __shared__ float sX[K2T * PAIRROW];   // 30720 B: X tile, pair-interleaved [k2][l][2]
    __shared__ float sx2[BLOCK_L];
    __shared__ float sw2[O_];

    const int tid = threadIdx.x;
    const int bi  = blockIdx.y;            // batch
    const int l0  = blockIdx.x * BLOCK_L;  // l-tile origin

    if (tid < 64) { sx2[tid] = 0.0f; sw2[tid] = 0.0f; }

    // Wave / fragment mapping (wave32)
    const int wave = tid >> 5;
    const int lane = tid & 31;
    const int half = lane >> 4;       // 0: lanes 0-15 (K offset 0), 1: lanes 16-31 (K offset +2)
    const int q    = lane & 15;       // M (A) / N (B, C, D)
    const int li   = wave & 3;        // l-subtile 0..3
    const int oi   = wave >> 2;       // o-pair 0..1
    const int lb   = li * 16;
    const int oA   = oi * 32;
    const int oB   = oA + 16;

    v8f acc0 = {0.f,0.f,0.f,0.f,0.f,0.f,0.f,0.f};
    v8f acc1 = {0.f,0.f,0.f,0.f,0.f,0.f,0.f,0.f};

    // Per-thread partial norms; this thread always touches columns (tid&15)*4 .. +3
    float px2[4] = {0.f, 0.f, 0.f, 0.f};
    float pw2[4] = {0.f, 0.f, 0.f, 0.f};

    // Per-wave LDS fragment base pointers: a v2f at +((kk/2)*PAIRROW) gives K = kk+2*half, kk+2*half+1
    const float* fX  = &sX[half * PAIRROW + 2 * (lb + q)];
    const float* fW0 = &sW[half * PAIRROW + 2 * (oA + q)];
    const float* fW1 = &sW[half * PAIRROW + 2 * (oB + q)];

    // Register staging buffers for one K-tile (double-buffer vs. LDS)
    float4 rwa[UNITS], rwb[UNITS], rxa[UNITS], rxb[UNITS];

    auto issue_loads = [&](int kt_) {
        const float* wt_ = w + (size_t)kt_ * KT * O_;
        const float* xt_ = x + ((size_t)bi * K_ + (size_t)kt_ * KT) * L_ + l0;
        #pragma unroll
        for (int i = 0; i < UNITS; ++i) {
            const int j  = tid + i * NTHREADS;
            const int k2 = j >> 4;        // pair-row 0..47
            const int c4 = j & 15;        // float4 column group
            rwa[i] = *(const float4*)(wt_ + (size_t)(2 * k2    ) * O_ + c4 * 4);
            rwb[i] = *(const float4*)(wt_ + (size_t)(2 * k2 + 1) * O_ + c4 * 4);
            rxa[i] = *(const float4*)(xt_ + (size_t)(2 * k2    ) * L_ + c4 * 4);
            rxb[i] = *(const float4*)(xt_ + (size_t)(2 * k2 + 1) * L_ + c4 * 4);
        }
    };

    auto store_tile = [&]() {
        #pragma unroll
        for (int i = 0; i < UNITS; ++i) {
            const int j  = tid + i * NTHREADS;
            const int k2 = j >> 4;
            const int c4 = j & 15;
            const float4 wa = rwa[i], wb = rwb[i], xa = rxa[i], xb = rxb[i];
            float* dW = &sW[k2 * PAIRROW + 8 * c4];
            float* dX = &sX[k2 * PAIRROW + 8 * c4];
            v2f t;
            t.x = wa.x; t.y = wb.x; *(v2f*)(dW + 0) = t;
            t.x = wa.y; t.y = wb.y; *(v2f*)(dW + 2) = t;
            t.x = wa.z; t.y = wb.z; *(v2f*)(dW + 4) = t;
            t.x = wa.w; t.y = wb.w; *(v2f*)(dW + 6) = t;
            t.x = xa.x; t.y = xb.x; *(v2f*)(dX + 0) = t;
            t.x = xa.y; t.y = xb.y; *(v2f*)(dX + 2) = t;
            t.x = xa.z; t.y = xb.z; *(v2f*)(dX + 4) = t;
            t.x = xa.w; t.y = xb.w; *(v2f*)(dX + 6) = t;
            pw2[0] += wa.x * wa.x + wb.x * wb.x;
            pw2[1] += wa.y * wa.y + wb.y * wb.y;
            pw2[2] += wa.z * wa.z + wb.z * wb.z;
            pw2[3] += wa.w * wa.w + wb.w * wb.w;
            px2[0] += xa.x * xa.x + xb.x * xb.x;
            px2[1] += xa.y * xa.y + xb.y * xb.y;
            px2[2] += xa.z * xa.z + xb.z * xb.z;
            px2[3] += xa.w * xa.w + xb.w * xb.w;
        }
    };

    issue_loads(0);

    for (int kt = 0; kt < K_ / KT; ++kt) {
        // drain in-flight global loads into LDS (+ norm partials)
        store_tile();
        __syncthreads();

        // kick off next tile's global loads; they fly underneath the 48 WMMAs below
        if (kt + 1 < K_ / KT) issue_loads(kt + 1);

        // ---- WMMA over this K-tile: 24 steps of K=4, software-pipelined b64 fragment loads ----
        v2f a  = *(const v2f*)(fX);
        v2f b0 = *(const v2f*)(fW0);
        v2f b1 = *(const v2f*)(fW1);
        #pragma unroll
        for (int kk = 0; kk < KT; kk += 4) {
            v2f an = a, b0n = b0, b1n = b1;
            if (kk + 4 < KT) {
                const int ro = ((kk + 4) >> 1) * PAIRROW;
                an  = *(const v2f*)(fX  + ro);
                b0n = *(const v2f*)(fW0 + ro);
                b1n = *(const v2f*)(fW1 + ro);
            }
            acc0 = __builtin_amdgcn_wmma_f32_16x16x4_f32(
                       false, a, false, b0, (short)0, acc0, false, false);
            acc1 = __builtin_amdgcn_wmma_f32_16x16x4_f32(
                       false, a, false, b1, (short)0, acc1, false, false);
            a = an; b0 = b0n; b1 = b1n;
        }
        __syncthreads();
    }

    // ---- reduce partial norms into LDS (ds_add_f32) ----
    {
        const int oc = (tid & 15) * 4;
        atomicAdd(&sx2[oc + 0], px2[0]);
        atomicAdd(&sx2[oc + 1], px2[1]);
        atomicAdd(&sx2[oc + 2], px2[2]);
        atomicAdd(&sx2[oc + 3], px2[3]);
        atomicAdd(&sw2[oc + 0], pw2[0]);
        atomicAdd(&sw2[oc + 1], pw2[1]);
        atomicAdd(&sw2[oc + 2], pw2[2]);
        atomicAdd(&sw2[oc + 3], pw2[3]);
    }
    __syncthreads();

    // ---- fused epilogue in registers, then repack via LDS for coalesced b128 stores ----
    const float g   = gamma[0];
    const float w2A = sw2[oA + q];
    const float w2B = sw2[oB + q];
    const float bA  = bias[oA + q];
    const float bB  = bias[oB + q];
    float* so = sW;   // reuse W staging buffer as [64][OROW] output tile

    #pragma unroll
    for (int r = 0; r < 8; ++r) {
        const int m = r + half * 8;    // C/D layout: VGPR r -> row r (lanes 0-15), row r+8 (lanes 16-31)
        const float x2v = sx2[lb + m];
        const float d0 = x2v + w2A - 2.0f * acc0[r];
        const float d1 = x2v + w2B - 2.0f * acc1[r];
        so[(lb + m) * OROW + oA + q] = __expf(-g * d0) + bA;
        so[(lb + m) * OROW + oB + q] = __expf(-g * d1) + bB;
    }
    __syncthreads();

    float* ob = out + ((size_t)bi * L_ + l0) * O_;
    #pragma unroll
    for (int i = 0; i < (BLOCK_L * 16) / NTHREADS; ++i) {   // 4 float4 per thread
        const int j   = tid + i * NTHREADS;
        const int row = j >> 4;
        const int c4  = j & 15;
        const float4 v = *(const float4*)&so[row * OROW + c4 * 4];
        *(float4*)(ob + (size_t)row * O_ + c4 * 4) = v;
    }
}

extern "C" void kernel_launch(void* const* d_in, const int* in_sizes, int n_in,
                              void* d_out, int out_size, void* d_ws, size_t ws_size,
                              hipStream_t stream) {
    (void)in_sizes; (void)n_in; (void)d_ws; (void)ws_size; (void)out_size;
    const float* x     = (const float*)d_in[0];   // [B, K, L]
    const float* w     = (const float*)d_in[1];   // [O,32,3,3] == [K,O] reinterpret
    const float* bias  = (const float*)d_in[2];   // [O]
    const float* gamma = (const float*)d_in[3];   // [1]
    float* out = (float*)d_out;                   // [B, L, O]

    dim3 grid(L_ / BLOCK_L, B_, 1);               // (256, 8)
    dim3 block(NTHREADS, 1, 1);
    gaussian_wmma_f32_kernel<<<grid, block, 0, stream>>>(x, w, bias, gamma, out);
}